// _RoIPooling_72464688218480
// MI455X (gfx1250) — compile-verified
//
#include <hip/hip_runtime.h>
#include <hip/hip_bf16.h>
#include <math.h>

// ---------------- problem constants (from the reference) ----------------
#define POOLED   7
#define C_TOTAL  256
#define H_FEAT   50
#define W_FEAT   50
#define SCALE    0.0625f

// ---------------- tiling ----------------
// Per-ROI union window is <= 18x18 (wh<=256 image px * 1/16 + rounding).
// Channels in chunks of 16 -> tile <= 16*18*18*4 = 20.7 KB; allocate 16*20*20
// and double-buffer: 2 * 25.6 KB = 51.2 KB static LDS (under 64 KB limit).
#define CCHUNK    16
#define NCHUNK    (C_TOTAL / CCHUNK)
#define MAXSPAN   20
#define TILE_ELEMS (CCHUNK * MAXSPAN * MAXSPAN)
#define NTHREADS  256

typedef unsigned int v4u __attribute__((ext_vector_type(4)));
typedef int          v4i __attribute__((ext_vector_type(4)));
typedef int          v8i __attribute__((ext_vector_type(8)));

// Sanity check only in the DEVICE pass (host pass lacks AMDGPU builtins).
#if defined(__HIP_DEVICE_COMPILE__)
#  if !__has_builtin(__builtin_amdgcn_tensor_load_to_lds)
#    error "gfx1250 TDM builtin __builtin_amdgcn_tensor_load_to_lds not available"
#  endif
#endif

static __device__ __forceinline__ unsigned rfl(unsigned v) {
  return (unsigned)__builtin_amdgcn_readfirstlane((int)v);
}

// Low 32 bits of the flat address of a __shared__ object == LDS byte offset
// (ISA §10.2).
static __device__ __forceinline__ unsigned lds_off_of(const void* p) {
  return (unsigned)(unsigned long long)p;
}

// Issue one TDM 3-D tile load: tile (wspan x hspan x CCHUNK) f32 from a
// (W=50, H*W=2500)-strided feature subvolume into a compact LDS tile.
// Descriptor per CDNA5 ISA §8.3-8.6. Wave-uniform values -> readfirstlane.
static __device__ __forceinline__ void tdm_issue(const float* gptr,
                                                 unsigned lds_byte,
                                                 int wspan, int hspan) {
#if defined(__HIP_DEVICE_COMPILE__)
  const unsigned long long ga = (unsigned long long)gptr;
  const unsigned ga_lo = rfl((unsigned)ga);
  const unsigned ga_hi = rfl((unsigned)(ga >> 32));
  const unsigned w  = rfl((unsigned)wspan);
  const unsigned h  = rfl((unsigned)hspan);
  const unsigned lo = rfl(lds_byte);

  // group0: [1:0] count=1 (valid), [63:32] lds_addr, [120:64] global_addr,
  //         [127:126] type=2 ("image")
  v4u g0 = {1u, lo, ga_lo, (ga_hi & 0x01FFFFFFu) | (2u << 30)};

  // group1: data_size=2 (4B); tensor dims == tile dims (no OOB path);
  // tensor_dim0_stride = W = 50, tensor_dim1_stride = H*W = 2500.
  v8i g1 = {
      (int)(2u << 16),                                     // data_size=4B
      (int)(w << 16),                                      // tensor_dim0 lo16
      (int)((w >> 16) | (h << 16)),                        // dim0 hi | dim1 lo
      (int)((h >> 16) | (w << 16)),                        // dim1 hi | tile_dim0
      (int)((h & 0xFFFFu) | ((unsigned)CCHUNK << 16)),     // tile_dim1 | tile_dim2
      (int)(W_FEAT),                                       // stride0 lo32 = 50
      (int)(((unsigned)(H_FEAT * W_FEAT) & 0xFFFFu) << 16),// stride0 hi=0 | stride1 lo16
      (int)((unsigned)(H_FEAT * W_FEAT) >> 16),            // stride1 hi32
  };
  v4i g2 = {CCHUNK, 0, 0, 0};                              // tensor_dim2
  v4i g3 = {0, 0, 0, 0};
#if defined(__clang_major__) && (__clang_major__ >= 23)
  v8i g4z = {0, 0, 0, 0, 0, 0, 0, 0};
  __builtin_amdgcn_tensor_load_to_lds(g0, g1, g2, g3, g4z, 0);
#else
  __builtin_amdgcn_tensor_load_to_lds(g0, g1, g2, g3, 0);
#endif
#else
  (void)gptr; (void)lds_byte; (void)wspan; (void)hspan;
#endif
}

__global__ void __launch_bounds__(NTHREADS)
roipool_tdm_kernel(const float* __restrict__ feats,
                   const float* __restrict__ rois,
                   float* __restrict__ out) {
  const int r   = blockIdx.x;
  const int tid = threadIdx.x;

  // ---- ROI geometry (uniform across the block; lowers to scalar FP) ----
  // jnp.round == round-half-to-even -> rintf (RNE is the default mode).
  const float x1f = rintf(rois[r * 5 + 1] * SCALE);
  const float y1f = rintf(rois[r * 5 + 2] * SCALE);
  const float x2f = rintf(rois[r * 5 + 3] * SCALE);
  const float y2f = rintf(rois[r * 5 + 4] * SCALE);
  const int b  = (int)rois[r * 5 + 0];
  const int x1 = (int)x1f, y1 = (int)y1f;
  const float roi_w = fmaxf(x2f - x1f + 1.0f, 1.0f);
  const float roi_h = fmaxf(y2f - y1f + 1.0f, 1.0f);
  const float bin_w = roi_w / 7.0f;   // keep exact division: matches reference
  const float bin_h = roi_h / 7.0f;

  // Union window over all 7x7 bins (bounds are monotone in p).
  int row_lo = min(max(y1, 0), H_FEAT);
  int row_hi = min(max((int)ceilf(7.0f * bin_h) + y1, 0), H_FEAT);
  int col_lo = min(max(x1, 0), W_FEAT);
  int col_hi = min(max((int)ceilf(7.0f * bin_w) + x1, 0), W_FEAT);
  row_hi = min(row_hi, row_lo + MAXSPAN);  // safety clamp (never binds for
  col_hi = min(col_hi, col_lo + MAXSPAN);  // this input distribution)
  const int hspan = row_hi - row_lo;
  const int wspan = col_hi - col_lo;

  float* outr = out + (size_t)r * C_TOTAL * POOLED * POOLED;

  if (hspan <= 0 || wspan <= 0) {  // whole ROI empty -> Caffe semantics: 0
    float4* o4 = (float4*)outr;    // 12544 floats, 16B-aligned
    const float4 z = {0.0f, 0.0f, 0.0f, 0.0f};
    for (int i = tid; i < (C_TOTAL * POOLED * POOLED) / 4; i += NTHREADS)
      o4[i] = z;
    return;
  }

  __shared__ float tile[2 * TILE_ELEMS];
  __shared__ int s_hs[POOLED], s_he[POOLED], s_ws[POOLED], s_we[POOLED];

  // Block-uniform per-bin boundaries, computed once (published by the
  // iteration-0 mid barrier below).
  if (tid < POOLED) {
    const float p0 = (float)tid, p1 = (float)(tid + 1);
    s_hs[tid] = min(max((int)floorf(p0 * bin_h) + y1, 0), H_FEAT);
    s_he[tid] = min(max((int)ceilf(p1 * bin_h) + y1, 0), H_FEAT);
    s_ws[tid] = min(max((int)floorf(p0 * bin_w) + x1, 0), W_FEAT);
    s_we[tid] = min(max((int)ceilf(p1 * bin_w) + x1, 0), W_FEAT);
  }

  const size_t chan_vol = (size_t)H_FEAT * W_FEAT;
  const size_t base0 =
      (((size_t)b * C_TOTAL) * H_FEAT + (size_t)row_lo) * W_FEAT + col_lo;

  // Prologue: wave 0 kicks off chunk 0. TDM ignores EXEC, but waves 1..7
  // take the uniformly-false branch and never reach the instruction.
  if (tid < 32)
    tdm_issue(feats + base0, lds_off_of(&tile[0]), wspan, hspan);

  for (int k = 0; k < NCHUNK; ++k) {
    const int buf = k & 1;
    float* tbuf = &tile[buf * TILE_ELEMS];

    if (tid < 32) {
      if (k + 1 < NCHUNK) {
        // Double-buffer: issue next tile, then wait until only it is in
        // flight (TENSORcnt is in-order per wave => chunk k has landed).
        tdm_issue(feats + base0 + (size_t)(k + 1) * CCHUNK * chan_vol,
                  lds_off_of(&tile[(buf ^ 1) * TILE_ELEMS]), wspan, hspan);
        __builtin_amdgcn_s_wait_tensorcnt(1);
      } else {
        __builtin_amdgcn_s_wait_tensorcnt(0);
      }
    }
    __syncthreads();  // publish TDM-written LDS (and, at k=0, bin tables)

    // ---- max-pool the 16 channels x 49 bins of this chunk from LDS ----
    const int c0 = k * CCHUNK;
    for (int i = tid; i < CCHUNK * 49; i += NTHREADS) {
      const int cl  = i / 49;
      const int bin = i - cl * 49;
      const int ph  = bin / 7;
      const int pw  = bin - ph * 7;

      int hs = s_hs[ph], he = s_he[ph];
      int ws = s_ws[pw], we = s_we[pw];
      const bool empty = (he <= hs) || (we <= ws);

      // keep reads inside the staged tile (no-ops for valid inputs)
      hs = max(hs, row_lo);  he = min(he, row_lo + hspan);
      ws = max(ws, col_lo);  we = min(we, col_lo + wspan);

      float m = -__builtin_inff();
      const float* tch = tbuf + cl * hspan * wspan;
      for (int h = hs; h < he; ++h) {
        const float* trow = tch + (h - row_lo) * wspan - col_lo;
        for (int w = ws; w < we; ++w) m = fmaxf(m, trow[w]);
      }
      outr[(c0 + cl) * 49 + bin] = empty ? 0.0f : m;
    }
    __syncthreads();  // tile buffer may be overwritten next iteration
  }
}

extern "C" void kernel_launch(void* const* d_in, const int* in_sizes, int n_in,
                              void* d_out, int out_size, void* d_ws, size_t ws_size,
                              hipStream_t stream) {
  (void)n_in; (void)out_size; (void)d_ws; (void)ws_size;
  const float* feats = (const float*)d_in[0];
  const float* rois  = (const float*)d_in[1];
  float* out         = (float*)d_out;
  const int R = in_sizes[1] / 5;  // rois are (R, 5)
  roipool_tdm_kernel<<<R, NTHREADS, 0, stream>>>(feats, rois, out);
}